// Extractor_87033217286585
// MI455X (gfx1250) — compile-verified
//
#include <hip/hip_runtime.h>

typedef __attribute__((ext_vector_type(2))) float v2f;
typedef __attribute__((ext_vector_type(8))) float v8f;

namespace {
constexpr int kH    = 384;
constexpr int kW    = 512;
constexpr int kNPix = kH * kW;   // 196608 = 768 * 256 exactly
constexpr int kNS   = 9;
constexpr int kVol  = 256;
constexpr int kBlk  = 256;
}

__global__ __launch_bounds__(kBlk) void Extractor_87033217286585_kernel(
    const float* __restrict__ depth,   // (1,H,W)
    const float* __restrict__ extr,    // (1,4,4)
    const float* __restrict__ intr,    // (1,3,3)
    const float* __restrict__ vol,     // (256,256,256)
    const float* __restrict__ vorig,   // (3,)
    float* __restrict__ out)
{
  __shared__ float sA[kBlk * 4];  // per-pixel [u*z, v*z, z, 1]
  __shared__ float sB[64];        // 4x16 B matrix, row-striped: sB[k*16+n]
  __shared__ float sW[kBlk * 4];  // per-pixel world xyz (+pad)

  const int tid  = threadIdx.x;
  const int lane = tid & 31;
  const int n15  = lane & 15;
  const int p    = blockIdx.x * kBlk + tid;   // flat pixel id
  const int row  = p / kW;
  const int col  = p - row * kW;

  // ---- Phase 0: build B = [ (E[:3,:3] @ inv(K))^T ; t^T ] padded to 4x16, in LDS.
  if (tid < 64) {
    float k00=intr[0],k01=intr[1],k02=intr[2];
    float k10=intr[3],k11=intr[4],k12=intr[5];
    float k20=intr[6],k21=intr[7],k22=intr[8];
    float c00 =  (k11*k22 - k12*k21);
    float c01 = -(k10*k22 - k12*k20);
    float c02 =  (k10*k21 - k11*k20);
    float id  = 1.0f / (k00*c00 + k01*c01 + k02*c02);
    float ki[3][3];
    ki[0][0] = c00*id;               ki[0][1] = (k02*k21-k01*k22)*id; ki[0][2] = (k01*k12-k02*k11)*id;
    ki[1][0] = c01*id;               ki[1][1] = (k00*k22-k02*k20)*id; ki[1][2] = (k02*k10-k00*k12)*id;
    ki[2][0] = c02*id;               ki[2][1] = (k01*k20-k00*k21)*id; ki[2][2] = (k00*k11-k01*k10)*id;
    float m[3][4];
    #pragma unroll
    for (int a = 0; a < 3; ++a) {
      #pragma unroll
      for (int b2 = 0; b2 < 3; ++b2)
        m[a][b2] = extr[a*4+0]*ki[0][b2] + extr[a*4+1]*ki[1][b2] + extr[a*4+2]*ki[2][b2];
      m[a][3] = extr[a*4+3];  // translation
    }
    const int kk = tid >> 4;     // K row 0..3
    const int nn = tid & 15;     // N col 0..15
    sB[kk*16 + nn] = (nn < 3) ? m[nn][kk] : 0.0f;
  }

  // ---- Phase 1: per-pixel homogeneous camera-plane point into LDS.
  const float z = depth[p];
  sA[tid*4 + 0] = (float)col * z;   // u*z
  sA[tid*4 + 1] = (float)row * z;   // v*z
  sA[tid*4 + 2] = z;
  sA[tid*4 + 3] = 1.0f;
  __syncthreads();

  // ---- Phase 2: two V_WMMA_F32_16X16X4_F32 per wave (32 pixels/wave).
  // A (16x4): VGPR0 = K{0,2}, VGPR1 = K{1,3} split by lane halves; M = lane&15.
  // B (4x16): VGPR0 = K{0,2}, VGPR1 = K{1,3} split by lane halves; N = lane&15.
  const int c0    = (lane < 16) ? 0 : 2;
  const int gbase = tid & ~31;            // wave's first local pixel
  v2f a0, a1, b;
  a0.x = sA[(gbase + n15)*4 + c0];
  a0.y = sA[(gbase + n15)*4 + c0 + 1];
  a1.x = sA[(gbase + 16 + n15)*4 + c0];
  a1.y = sA[(gbase + 16 + n15)*4 + c0 + 1];
  b.x  = sB[(c0    )*16 + n15];
  b.y  = sB[(c0 + 1)*16 + n15];
  v8f cz = {0.f,0.f,0.f,0.f,0.f,0.f,0.f,0.f};
  v8f d0 = __builtin_amdgcn_wmma_f32_16x16x4_f32(false, a0, false, b, (short)0, cz, false, false);
  v8f d1 = __builtin_amdgcn_wmma_f32_16x16x4_f32(false, a1, false, b, (short)0, cz, false, false);

  // D layout: VGPR g -> M = g (lanes 0-15) / g+8 (lanes 16-31), N = lane&15.
  // Only N columns 0..2 (world x,y,z) are meaningful.
  if (n15 < 3) {
    const int hi = (lane >> 4) << 3;   // 0 or 8
    #pragma unroll
    for (int g = 0; g < 8; ++g) {
      sW[(gbase + hi + g     )*4 + n15] = d0[g];
      sW[(gbase + hi + g + 16)*4 + n15] = d1[g];
    }
  }
  __syncthreads();

  // ---- Phase 3: rays, corner selection (geometry-only), single gather/sample.
  const float wx = sW[tid*4+0], wy = sW[tid*4+1], wz = sW[tid*4+2];
  const float ex = extr[3], ey = extr[7], ez = extr[11];   // eye = E[:3,3]
  float dx = wx - ex, dy = wy - ey, dz = wz - ez;
  const float rinv = 1.0f / fmaxf(sqrtf(dx*dx + dy*dy + dz*dz), 1e-12f);
  dx *= rinv; dy *= rinv; dz *= rinv;
  const float vx = wx - vorig[0], vy = wy - vorig[1], vz = wz - vorig[2];

  float* __restrict__ fusion = out;
  float* __restrict__ selidx = out + (size_t)kNPix * kNS;        // + N*9
  float* __restrict__ rpts   = out + (size_t)kNPix * kNS * 4;    // + N*9 + N*27
  float* __restrict__ rdirs  = out + (size_t)kNPix * kNS * 7;    // + N*9 + 2*N*27

  #pragma unroll
  for (int s = 0; s < kNS; ++s) {
    const float st = (float)(s - 4);                 // (s - (n-1)/2) * STEP
    const float px = vx + st*dx, py = vy + st*dy, pz = vz + st*dz;
    const float bx = floorf(px), by = floorf(py), bz = floorf(pz);
    const float fx = px - bx, fy = py - by, fz = pz - bz;
    const float gx = 1.0f - fx, gy = 1.0f - fy, gz = 1.0f - fz;
    // argmin over 8 corners of |d|x*|d|y*|d|z; first-min wins (matches jnp.argmin).
    float best = 3.4e38f; int sel = 0;
    #pragma unroll
    for (int ci = 0; ci < 8; ++ci) {
      const float d = ((ci & 4) ? gx : fx) * ((ci & 2) ? gy : fy) * ((ci & 1) ? gz : fz);
      if (d < best) { best = d; sel = ci; }
    }
    const int ix = (int)bx + ((sel >> 2) & 1);
    const int iy = (int)by + ((sel >> 1) & 1);
    const int iz = (int)bz + (sel & 1);
    const bool ok = ((unsigned)ix < (unsigned)kVol) &
                    ((unsigned)iy < (unsigned)kVol) &
                    ((unsigned)iz < (unsigned)kVol);
    float val = 0.0f;
    if (ok) val = vol[((size_t)ix * kVol + iy) * kVol + iz];
    const size_t q = (size_t)p * kNS + s;
    fusion[q] = val;
    selidx[q*3+0] = (float)ix;
    selidx[q*3+1] = (float)iy;
    selidx[q*3+2] = (float)iz;
    rpts[q*3+0] = px;
    rpts[q*3+1] = py;
    rpts[q*3+2] = pz;
  }
  rdirs[(size_t)p*3+0] = dx;
  rdirs[(size_t)p*3+1] = dy;
  rdirs[(size_t)p*3+2] = dz;
}

extern "C" void kernel_launch(void* const* d_in, const int* in_sizes, int n_in,
                              void* d_out, int out_size, void* d_ws, size_t ws_size,
                              hipStream_t stream) {
  (void)in_sizes; (void)n_in; (void)out_size; (void)d_ws; (void)ws_size;
  const float* depth = (const float*)d_in[0];
  const float* extr  = (const float*)d_in[1];
  const float* intr  = (const float*)d_in[2];
  const float* vol   = (const float*)d_in[3];
  const float* vorig = (const float*)d_in[4];
  Extractor_87033217286585_kernel<<<kNPix / kBlk, kBlk, 0, stream>>>(
      depth, extr, intr, vol, vorig, (float*)d_out);
}